// CRF_83580063580574
// MI455X (gfx1250) — compile-verified
//
#include <hip/hip_runtime.h>
#include <hip/hip_bf16.h>
#include <cfloat>

// Problem constants (from reference setup_inputs)
#define BATCH  2
#define NPTS   4096
#define NCLS   32
#define NFEAT  256
#define KNN    16
#define CAP    64            // per-node symmetric-adjacency capacity (deg ~32 expected)
#define BN     (BATCH * NPTS)

typedef float v2f __attribute__((ext_vector_type(2)));
typedef float v8f __attribute__((ext_vector_type(8)));

__device__ __forceinline__ float softplusf(float x) { return log1pf(expf(x)); }

__device__ __forceinline__ float waveReduceSum(float v) {
#pragma unroll
  for (int off = 16; off; off >>= 1) v += __shfl_xor(v, off, 32);
  return v;
}
__device__ __forceinline__ float waveReduceMax(float v) {
#pragma unroll
  for (int off = 16; off; off >>= 1) v = fmaxf(v, __shfl_xor(v, off, 32));
  return v;
}

// ---------------------------------------------------------------------------
// 1) centers (float4: x,y,z, w = valid ? |c|^2 : -1) + L2-normalized features
//    one wave per node
// ---------------------------------------------------------------------------
__global__ void crf_prep(const float* __restrict__ rois,
                         const float* __restrict__ feats,
                         float4* __restrict__ centers,
                         float* __restrict__ featsN) {
  const int lane = threadIdx.x & 31;
  const int node = blockIdx.x * 4 + (threadIdx.x >> 5);
  const float* f = feats + (size_t)node * NFEAT;
  float fv[8];
  float ss = 0.f;
#pragma unroll
  for (int t = 0; t < 8; ++t) { fv[t] = f[lane + 32 * t]; ss += fv[t] * fv[t]; }
  ss = waveReduceSum(ss);
  const float inv = 1.f / fmaxf(sqrtf(ss), 1e-6f);
  float* fo = featsN + (size_t)node * NFEAT;
#pragma unroll
  for (int t = 0; t < 8; ++t) fo[lane + 32 * t] = fv[t] * inv;
  if (lane == 0) {
    const float* r = rois + (size_t)node * 6;
    const float r0 = r[0], r1 = r[1], r2 = r[2], r3 = r[3], r4 = r[4], r5 = r[5];
    const float cx = (r0 + r3) * 0.5f, cy = (r1 + r4) * 0.5f, cz = (r2 + r5) * 0.5f;
    const float sq = cx * cx + cy * cy + cz * cz;
    const bool valid =
        (fabsf(r0) + fabsf(r1) + fabsf(r2) + fabsf(r3) + fabsf(r4) + fabsf(r5)) > 0.f;
    centers[node] = make_float4(cx, cy, cz, valid ? sq : -1.f);
  }
}

// ---------------------------------------------------------------------------
// 2) kNN threshold: thr[node] = 16th smallest masked dist_sq (FLT_MAX if <16
//    candidates, -1 if node invalid). Symmetrized adjacency is then exactly
//    dist_sq(n,m) <= max(thr_n, thr_m). One thread per node, register top-16.
// ---------------------------------------------------------------------------
__global__ void crf_knn_thresh(const float4* __restrict__ centers,
                               float* __restrict__ thr) {
  const int node = blockIdx.x * blockDim.x + threadIdx.x;
  const int b = node / NPTS, n = node % NPTS;
  const float4 cn = centers[node];
  if (cn.w < 0.f) { thr[node] = -1.f; return; }
  const float sqn = cn.w;
  float best[KNN];
#pragma unroll
  for (int i = 0; i < KNN; ++i) best[i] = FLT_MAX;
  float curMax = FLT_MAX;
  int maxPos = 0;
  const float4* cb = centers + b * NPTS;
  for (int m = 0; m < NPTS; ++m) {
    const float4 cm = cb[m];
    if (m == n || cm.w < 0.f) continue;
    const float dot = cn.x * cm.x + cn.y * cm.y + cn.z * cm.z;
    const float d = fmaxf(sqn + cm.w - 2.f * dot, 0.f);
    if (d < curMax) {
#pragma unroll
      for (int i = 0; i < KNN; ++i)
        if (i == maxPos) best[i] = d;
      curMax = best[0]; maxPos = 0;
#pragma unroll
      for (int i = 1; i < KNN; ++i)
        if (best[i] > curMax) { curMax = best[i]; maxPos = i; }
    }
  }
  thr[node] = curMax;
}

// ---------------------------------------------------------------------------
// 3) sparse edge build: one wave per node. Lanes scan candidates, ballot the
//    edges, then all 32 lanes cooperatively compute the 256-d cosine per edge.
//    Weights row-normalized in place. Edge count ~2*KNN.
// ---------------------------------------------------------------------------
__global__ void crf_edges(const float4* __restrict__ centers,
                          const float* __restrict__ thr,
                          const float* __restrict__ featsN,
                          const float* __restrict__ raw_sigma,
                          int* __restrict__ eidx,
                          float* __restrict__ ew,
                          int* __restrict__ deg) {
  const int lane = threadIdx.x & 31;
  const int node = blockIdx.x * 4 + (threadIdx.x >> 5);
  const int b = node / NPTS, n = node % NPTS;
  const float4 cn = centers[node];
  if (cn.w < 0.f) { if (lane == 0) deg[node] = 0; return; }
  const float thrN = thr[node];
  const float sigma = fmaxf(softplusf(raw_sigma[0]), 1e-6f);
  const float gscale = -0.5f / (sigma * sigma);
  // cache own normalized feature row in registers
  const float* fn = featsN + (size_t)node * NFEAT;
  float fnr[8];
#pragma unroll
  for (int t = 0; t < 8; ++t) fnr[t] = fn[lane + 32 * t];

  const float4* cb = centers + b * NPTS;
  const float* tb = thr + b * NPTS;
  int cnt = 0;
  float wsum = 0.f;
  for (int base = 0; base < NPTS; base += 32) {
    const int m = base + lane;
    const float4 cm = cb[m];
    bool ok = (m != n) && (cm.w >= 0.f);
    float d = 0.f;
    if (ok) {
      const float dot = cn.x * cm.x + cn.y * cm.y + cn.z * cm.z;
      d = fmaxf(cn.w + cm.w - 2.f * dot, 0.f);
      ok = d <= fmaxf(thrN, tb[m]);
    }
    unsigned long long mask = __ballot(ok);
    while (mask && cnt < CAP) {
      const int j = __ffsll(mask) - 1;
      mask &= mask - 1;
      const int mj = base + j;
      const float dj = __shfl(d, j, 32);
      const float* fm = featsN + ((size_t)(b * NPTS + mj)) * NFEAT;
      float p = 0.f;
#pragma unroll
      for (int t = 0; t < 8; ++t) p += fnr[t] * fm[lane + 32 * t];
      p = waveReduceSum(p);  // cosine, broadcast to all lanes
      const float aff = fminf(fmaxf((p + 1.f) * 0.5f, 0.f), 1.f);
      const float w = expf(dj * gscale) * aff;
      if (lane == 0) { eidx[node * CAP + cnt] = mj; ew[node * CAP + cnt] = w; }
      wsum += w;
      ++cnt;
    }
  }
  const float inv = 1.f / fmaxf(wsum, 1e-6f);
  for (int e = lane; e < cnt; e += 32) ew[node * CAP + e] *= inv;
  if (lane == 0) deg[node] = cnt;
}

// ---------------------------------------------------------------------------
// 4) softmax over classes: one wave per node, lane = class
// ---------------------------------------------------------------------------
__global__ void crf_softmax(const float* __restrict__ in, float* __restrict__ q) {
  const int lane = threadIdx.x & 31;
  const int node = blockIdx.x * 4 + (threadIdx.x >> 5);
  const float x = in[(size_t)node * NCLS + lane];
  const float m = waveReduceMax(x);
  const float e = expf(x - m);
  const float s = waveReduceSum(e);
  q[(size_t)node * NCLS + lane] = e / s;
}

// ---------------------------------------------------------------------------
// 5) mean-field step: one wave per 16-node tile.
//    Phase A: sparse qn = K·q into LDS (lane = class).
//    Phase B: pairwise = qn(16x32) x compat(32x32) via V_WMMA_F32_16X16X4_F32
//             (fp32-exact); compat synthesized in VALU, zero memory traffic.
//    refined = logits - smoothness * pairwise fused into the D-layout store.
// ---------------------------------------------------------------------------
__global__ void crf_meanfield(const float* __restrict__ q,
                              const int* __restrict__ eidx,
                              const float* __restrict__ ew,
                              const int* __restrict__ deg,
                              const float* __restrict__ logits,
                              const float* __restrict__ raw_smooth,
                              float* __restrict__ out) {
  __shared__ float qn_s[4][16 * NCLS];  // 8 KB
  const int lane = threadIdx.x & 31;
  const int wave = threadIdx.x >> 5;
  const int tile = blockIdx.x * 4 + wave;
  const int n0 = tile * 16;
  const int b = n0 / NPTS;
  float* qn = qn_s[wave];

  // Phase A: per-node sparse aggregation, lane = class
  for (int i = 0; i < 16; ++i) {
    const int nd = n0 + i;
    const int dg = deg[nd];
    const int* ei = eidx + (size_t)nd * CAP;
    const float* wv = ew + (size_t)nd * CAP;
    float acc = 0.f;
    for (int e = 0; e < dg; ++e) {
      const int m = ei[e];
      const float w = wv[e];
      acc += w * q[((size_t)(b * NPTS + m)) * NCLS + lane];
    }
    qn[i * NCLS + lane] = acc;
  }
  __syncthreads();

  const float smooth = softplusf(raw_smooth[0]);
  const int row = lane & 15;        // A: M row (lanes 0-15 / 16-31)
  const int khalf = lane >> 4;      // A/B: K sub-pair select
  const float cnorm = 1.0f / 961.0f;  // 1 / (C-1)^2

#pragma unroll
  for (int t = 0; t < 2; ++t) {     // two 16-wide output column tiles
    const int c0 = t * 16;
    const int col = c0 + row;       // B column / D column for this lane
    v8f acc = {0.f, 0.f, 0.f, 0.f, 0.f, 0.f, 0.f, 0.f};
#pragma unroll
    for (int k = 0; k < NCLS; k += 4) {
      const int kk = k + khalf * 2;
      v2f a;
      a.x = qn[row * NCLS + kk];
      a.y = qn[row * NCLS + kk + 1];
      v2f bb;
      const float d0 = (float)(kk - col);
      const float d1 = (float)(kk + 1 - col);
      bb.x = d0 * d0 * cnorm;
      bb.y = d1 * d1 * cnorm;
      acc = __builtin_amdgcn_wmma_f32_16x16x4_f32(
          /*neg_a=*/false, a, /*neg_b=*/false, bb,
          /*c_mod=*/(short)0, acc, /*reuse_a=*/false, /*reuse_b=*/false);
    }
    // D layout: VGPR r -> M=r (lanes 0-15) / M=r+8 (lanes 16-31), N = col
    const int mbase = khalf * 8;
#pragma unroll
    for (int r = 0; r < 8; ++r) {
      const size_t off = (size_t)(n0 + mbase + r) * NCLS + col;
      out[off] = logits[off] - smooth * acc[r];
    }
  }
}

// ---------------------------------------------------------------------------
// launcher
// ---------------------------------------------------------------------------
extern "C" void kernel_launch(void* const* d_in, const int* in_sizes, int n_in,
                              void* d_out, int out_size, void* d_ws, size_t ws_size,
                              hipStream_t stream) {
  const float* logits     = (const float*)d_in[0];
  const float* rois       = (const float*)d_in[1];
  const float* feats      = (const float*)d_in[2];
  const float* raw_sigma  = (const float*)d_in[3];
  const float* raw_smooth = (const float*)d_in[4];
  float* out = (float*)d_out;

  // workspace partition (~13.3 MB total)
  char* ws = (char*)d_ws;
  float4* centers = (float4*)ws;            ws += (size_t)BN * sizeof(float4);
  float* thr      = (float*)ws;             ws += (size_t)BN * sizeof(float);
  int*   deg      = (int*)ws;               ws += (size_t)BN * sizeof(int);
  float* featsN   = (float*)ws;             ws += (size_t)BN * NFEAT * sizeof(float);
  int*   eidx     = (int*)ws;               ws += (size_t)BN * CAP * sizeof(int);
  float* ew       = (float*)ws;             ws += (size_t)BN * CAP * sizeof(float);
  float* q        = (float*)ws;             ws += (size_t)BN * NCLS * sizeof(float);

  crf_prep<<<BN / 4, 128, 0, stream>>>(rois, feats, centers, featsN);
  crf_knn_thresh<<<BN / 256, 256, 0, stream>>>(centers, thr);
  crf_edges<<<BN / 4, 128, 0, stream>>>(centers, thr, featsN, raw_sigma, eidx, ew, deg);

  crf_softmax<<<BN / 4, 128, 0, stream>>>(logits, q);
  for (int it = 0; it < 5; ++it) {
    crf_meanfield<<<(BN / 16) / 4, 128, 0, stream>>>(q, eidx, ew, deg, logits,
                                                     raw_smooth, out);
    if (it < 4) crf_softmax<<<BN / 4, 128, 0, stream>>>(out, q);
  }
}